// GNNEncoder_81363860455754
// MI455X (gfx1250) — compile-verified
//
#include <hip/hip_runtime.h>

#define NN 100000
#define NE 1600000
#define HID 128
#define NG 2048
#define EPSV 1e-5f

typedef __attribute__((ext_vector_type(2))) float v2f;
typedef __attribute__((ext_vector_type(8))) float v8f;

// Guaranteed hardware f32 atomic add (no-return form, STOREcnt-tracked).
// Avoids any chance of a CAS-loop fallback on the dominant scatter path.
__device__ __forceinline__ void atomAddF(float* p, float v) {
  asm volatile("global_atomic_add_f32 %0, %1, off" : : "v"(p), "v"(v) : "memory");
}

__global__ void fill_kernel(float* __restrict__ p, float v, int n) {
  int i = blockIdx.x * blockDim.x + threadIdx.x;
  if (i < n) p[i] = v;
}

// deg[c] += 1 for every edge destination (self-loop contribution pre-filled as 1.0)
__global__ void deg_accum_kernel(const int* __restrict__ col, float* __restrict__ deg, int nE) {
  int e = blockIdx.x * blockDim.x + threadIdx.x;
  if (e < nE) atomAddF(&deg[col[e]], 1.0f);
}

__global__ void rsqrt_kernel(float* __restrict__ p, int n) {
  int i = blockIdx.x * blockDim.x + threadIdx.x;
  if (i < n) p[i] = rsqrtf(fmaxf(p[i], 1.0f));
}

// Layer-1 GEMM: K=6, plain FMA (too small for matrix cores)
__global__ void gemm6_kernel(const float* __restrict__ x, const float* __restrict__ W,
                             float* __restrict__ out, int nNodes) {
  int idx = blockIdx.x * blockDim.x + threadIdx.x;
  if (idx >= nNodes * HID) return;
  int node = idx >> 7, j = idx & (HID - 1);
  const float* xr = x + node * 6;
  float acc = 0.f;
#pragma unroll
  for (int k = 0; k < 6; k++) acc = fmaf(xr[k], W[k * HID + j], acc);
  out[idx] = acc;
}

__global__ void init_bias_kernel(float* __restrict__ agg, const float* __restrict__ b, int n) {
  int i = blockIdx.x * blockDim.x + threadIdx.x;
  if (i < n) agg[i] = b[i & (HID - 1)];
}

// 128x128 transpose: Wt[n][k] = W[k][n].  Makes WMMA B-fragment loads
// (B[k][n], B[k+1][n]) contiguous -> single global_load_b64 per k-step.
__global__ void transpose128_kernel(const float* __restrict__ W, float* __restrict__ Wt) {
  int i = blockIdx.x * blockDim.x + threadIdx.x;   // 16384 threads
  int k = i >> 7, n = i & (HID - 1);
  Wt[(size_t)n * HID + k] = W[(size_t)k * HID + n];
}

// One wave32 per (edge or self-loop): gather 128 floats of hw[row], scale by
// dinv[row]*dinv[col], scatter-add into agg[col].  L2-resident working set.
__global__ __launch_bounds__(256) void aggregate_kernel(const float4* __restrict__ hw,
                                                        float* __restrict__ agg,
                                                        const int* __restrict__ ei,
                                                        const float* __restrict__ dinv) {
  int gw = (blockIdx.x * blockDim.x + threadIdx.x) >> 5;
  int lane = threadIdx.x & 31;
  const int total = NE + NN;
  if (gw >= total) return;
  int r, c;
  if (gw < NE) { r = ei[gw]; c = ei[NE + gw]; }
  else         { r = c = gw - NE; }
  float norm = dinv[r] * dinv[c];
  float4 v = hw[(size_t)r * (HID / 4) + lane];
  float* dst = agg + (size_t)c * HID + lane * 4;
  atomAddF(dst + 0, v.x * norm);
  atomAddF(dst + 1, v.y * norm);
  atomAddF(dst + 2, v.z * norm);
  atomAddF(dst + 3, v.w * norm);
}

__global__ void bnrelu_kernel(float* __restrict__ h, const float* __restrict__ g,
                              const float* __restrict__ beta, const float* __restrict__ m,
                              const float* __restrict__ v, int n) {
  int i = blockIdx.x * blockDim.x + threadIdx.x;
  if (i >= n) return;
  int j = i & (HID - 1);
  float val = (h[i] - m[j]) * rsqrtf(v[j] + EPSV) * g[j] + beta[j];
  h[i] = fmaxf(val, 0.f);
}

// fp32 WMMA GEMM: Out[m0:m0+16, :] = H[m0:m0+16, 0:128] @ W[0:128, 0:128]
// Block = 8 waves; wave w owns N-tile [16w, 16w+16).  A tile staged in LDS
// (stride 130 floats -> conflict-free ds_load_2addr_b64).  B read from the
// pre-transposed Wt so each lane's (k, k+1) pair is one global_load_b64.
// K consumed 4 at a time with V_WMMA_F32_16X16X4_F32 (ISA 7.12.2 layouts).
__global__ __launch_bounds__(256) void gemm_wmma_kernel(const float* __restrict__ H,
                                                        const float* __restrict__ Wt,
                                                        float* __restrict__ Out) {
  __shared__ float As[16 * 130];
  const int m0 = blockIdx.x * 16;
  const int wave = threadIdx.x >> 5;
  const int lane = threadIdx.x & 31;
  for (int i = threadIdx.x; i < 16 * HID; i += 256) {
    int r = i >> 7, k = i & (HID - 1);
    As[r * 130 + k] = H[(size_t)(m0 + r) * HID + k];
  }
  __syncthreads();
  const int n0 = wave * 16;
  const int hl = lane & 15;            // row (A) / col (B,D) within tile
  const int kOff = (lane >> 4) * 2;    // lanes 0-15: K+0/1, lanes 16-31: K+2/3
  const float* wrow = Wt + (size_t)(n0 + hl) * HID;   // Wt[n][*]
  v8f c = {};
  for (int k0 = 0; k0 < HID; k0 += 4) {
    v2f a = *(const v2f*)&As[hl * 130 + k0 + kOff];        // 8B aligned
    v2f b = *(const v2f*)&wrow[k0 + kOff];                  // 8B aligned
    c = __builtin_amdgcn_wmma_f32_16x16x4_f32(false, a, false, b, (short)0, c, false, false);
  }
#pragma unroll
  for (int r = 0; r < 8; r++) {
    int m = (lane < 16) ? r : (r + 8);
    Out[(size_t)(m0 + m) * HID + n0 + hl] = c[r];
  }
}

__global__ void pool_sum_kernel(const float* __restrict__ h, const int* __restrict__ batch,
                                float* __restrict__ out, float* __restrict__ cnt, int n) {
  int i = blockIdx.x * blockDim.x + threadIdx.x;
  if (i >= n) return;
  int node = i >> 7, j = i & (HID - 1);
  int b = batch[node];
  atomAddF(&out[(size_t)b * HID + j], h[i]);
  if (j == 0) atomAddF(&cnt[b], 1.0f);
}

__global__ void pool_div_kernel(float* __restrict__ out, const float* __restrict__ cnt, int n) {
  int i = blockIdx.x * blockDim.x + threadIdx.x;
  if (i >= n) return;
  out[i] /= fmaxf(cnt[i >> 7], 1.0f);
}

extern "C" void kernel_launch(void* const* d_in, const int* in_sizes, int n_in,
                              void* d_out, int out_size, void* d_ws, size_t ws_size,
                              hipStream_t stream) {
  (void)in_sizes; (void)n_in; (void)out_size; (void)ws_size;
  const float* x     = (const float*)d_in[0];
  const int*   ei    = (const int*)d_in[1];      // [2, NE]: rows then cols
  const int*   batch = (const int*)d_in[3];
  const float* W1 = (const float*)d_in[4];  const float* b1 = (const float*)d_in[5];
  const float* W2 = (const float*)d_in[6];  const float* b2 = (const float*)d_in[7];
  const float* W3 = (const float*)d_in[8];  const float* b3 = (const float*)d_in[9];
  const float* g1  = (const float*)d_in[10]; const float* be1 = (const float*)d_in[11];
  const float* m1  = (const float*)d_in[12]; const float* v1  = (const float*)d_in[13];
  const float* g2  = (const float*)d_in[14]; const float* be2 = (const float*)d_in[15];
  const float* m2  = (const float*)d_in[16]; const float* v2  = (const float*)d_in[17];
  const float* g3  = (const float*)d_in[18]; const float* be3 = (const float*)d_in[19];
  const float* m3  = (const float*)d_in[20]; const float* v3  = (const float*)d_in[21];
  float* out = (float*)d_out;

  const size_t NH64 = (size_t)NN * HID;
  float* bufA = (float*)d_ws;          // 51.2 MB
  float* bufB = bufA + NH64;           // 51.2 MB
  float* dinv = bufB + NH64;           // 400 KB (deg, then rsqrt in place)
  float* cnt  = dinv + NN;             // 8 KB
  float* wt   = cnt + NG;              // 64 KB transposed weight scratch

  const int T = 256;
  const int NH = NN * HID;                       // 12.8M, fits int
  const int gNH = (NH + T - 1) / T;
  const int gN  = (NN + T - 1) / T;
  const long aggThreads = (long)(NE + NN) * 32;
  const int gAgg = (int)((aggThreads + T - 1) / T);
  const int gW = (HID * HID) / T;                // 64 blocks

  // symmetric normalization: deg = 1 (self loop) + indegree; dinv = rsqrt
  fill_kernel<<<gN, T, 0, stream>>>(dinv, 1.0f, NN);
  deg_accum_kernel<<<(NE + T - 1) / T, T, 0, stream>>>(ei + NE, dinv, NE);
  rsqrt_kernel<<<gN, T, 0, stream>>>(dinv, NN);

  // ---- layer 1 ----
  gemm6_kernel<<<gNH, T, 0, stream>>>(x, W1, bufA, NN);
  init_bias_kernel<<<gNH, T, 0, stream>>>(bufB, b1, NH);
  aggregate_kernel<<<gAgg, T, 0, stream>>>((const float4*)bufA, bufB, ei, dinv);
  bnrelu_kernel<<<gNH, T, 0, stream>>>(bufB, g1, be1, m1, v1, NH);

  // ---- layer 2 ----
  transpose128_kernel<<<gW, T, 0, stream>>>(W2, wt);
  gemm_wmma_kernel<<<NN / 16, T, 0, stream>>>(bufB, wt, bufA);
  init_bias_kernel<<<gNH, T, 0, stream>>>(bufB, b2, NH);
  aggregate_kernel<<<gAgg, T, 0, stream>>>((const float4*)bufA, bufB, ei, dinv);
  bnrelu_kernel<<<gNH, T, 0, stream>>>(bufB, g2, be2, m2, v2, NH);

  // ---- layer 3 ----
  transpose128_kernel<<<gW, T, 0, stream>>>(W3, wt);
  gemm_wmma_kernel<<<NN / 16, T, 0, stream>>>(bufB, wt, bufA);
  init_bias_kernel<<<gNH, T, 0, stream>>>(bufB, b3, NH);
  aggregate_kernel<<<gAgg, T, 0, stream>>>((const float4*)bufA, bufB, ei, dinv);
  bnrelu_kernel<<<gNH, T, 0, stream>>>(bufB, g3, be3, m3, v3, NH);

  // ---- global mean pool ----
  fill_kernel<<<(NG * HID + T - 1) / T, T, 0, stream>>>(out, 0.0f, NG * HID);
  fill_kernel<<<(NG + T - 1) / T, T, 0, stream>>>(cnt, 0.0f, NG);
  pool_sum_kernel<<<gNH, T, 0, stream>>>(bufB, batch, out, cnt, NH);
  pool_div_kernel<<<(NG * HID + T - 1) / T, T, 0, stream>>>(out, cnt, NG * HID);
}